// HEALPixPad_42356967473531
// MI455X (gfx1250) — compile-verified
//
#include <hip/hip_runtime.h>

// ---------------------------------------------------------------------------
// HEALPix pad (p=2) for x: (B=2, F=12, C=32, 256, 256) f32 -> (..., 260, 260)
// Memory-bound: ~409MB traffic, roofline ~17.6us @ 23.3 TB/s.
// Interior copy uses CDNA5 async global<->LDS DMA path (ASYNCcnt).
// ---------------------------------------------------------------------------

#ifndef __has_builtin
#define __has_builtin(x) 0
#endif

#define HAVE_ASYNC_LDS (__has_builtin(__builtin_amdgcn_global_load_async_to_lds_b128) && \
                        __has_builtin(__builtin_amdgcn_global_store_async_from_lds_b64))

// Builtin parameter types (from hipcc diagnostics): GCC-vector-of-int pointees,
// AS1 (global / __device__) for param 0, AS3 (LDS) for param 1.
typedef int v4i __attribute__((vector_size(16)));
typedef int v2i __attribute__((vector_size(8)));
typedef __attribute__((address_space(1))) v4i* gv4p;
typedef __attribute__((address_space(3))) v4i* lv4p;
typedef __attribute__((address_space(1))) v2i* gv2p;
typedef __attribute__((address_space(3))) v2i* lv2p;

static constexpr int N  = 256;   // face size
static constexpr int P  = 2;     // pad
static constexpr int HO = 260;   // padded size
static constexpr int ROWS = 16;  // rows per interior block
static constexpr int CHUNK_F = ROWS * N;   // 4096 floats = 16KB LDS

// ------------------------------- interior ----------------------------------
__global__ __launch_bounds__(256)
void hp_interior(const float* __restrict__ in, float* __restrict__ out) {
  const int t  = threadIdx.x;          // 0..255
  const int rg = blockIdx.x;           // 0..15 row-group
  const int pl = blockIdx.y;           // 0..planes-1  (b*12*32 flat plane)
  const size_t in_base  = ((size_t)pl << 16) + (size_t)rg * CHUNK_F;   // floats
  const size_t out_base = (size_t)pl * (HO * HO);                      // floats

#if HAVE_ASYNC_LDS
  __shared__ float buf[CHUNK_F];
  // Phase 1: coalesced async loads global -> LDS (16B per lane per op)
#pragma unroll
  for (int k = 0; k < 4; ++k) {
    const int g4 = (k * 256 + t) * 4;                 // float index in chunk
    __builtin_amdgcn_global_load_async_to_lds_b128(
        (gv4p)(in + in_base + g4), (lv4p)&buf[g4], 0, 0);
  }
#if __has_builtin(__builtin_amdgcn_s_wait_asynccnt)
  __builtin_amdgcn_s_wait_asynccnt(0);
#else
  asm volatile("s_wait_asynccnt 0x0" ::: "memory");
#endif
  __syncthreads();   // stores read LDS written by other waves' loads
  // Phase 2: async stores LDS -> global. Output rows are only 8B-aligned
  // ((y+2)*260+2 floats), so use b64 units. 32 lanes * 8B = 256B contiguous.
#pragma unroll
  for (int m = 0; m < 8; ++m) {
    const int u = m * 256 + t;          // 8-byte unit within chunk [0,2048)
    const int r = u >> 7;               // row in chunk (128 units per row)
    const int c = (u & 127) * 2;        // column (floats)
    float* dst = out + out_base + (size_t)(rg * ROWS + r + P) * HO + P + c;
    __builtin_amdgcn_global_store_async_from_lds_b64(
        (gv2p)dst, (lv2p)&buf[u * 2], 0, 0);
  }
  // S_ENDPGM performs implicit wait-idle (covers outstanding ASYNCcnt).
#else
  // Fallback: VGPR copy, float4 loads (16B aligned), 2x float2 stores.
#pragma unroll
  for (int k = 0; k < 4; ++k) {
    const int g = (k * 256 + t) * 4;    // float index in chunk
    const float4 v = *(const float4*)(in + in_base + g);
    const int r = g >> 8, c = g & 255;
    float* dst = out + out_base + (size_t)(rg * ROWS + r + P) * HO + P + c;
    ((float2*)dst)[0] = make_float2(v.x, v.y);
    ((float2*)dst)[1] = make_float2(v.z, v.w);
  }
#endif
}

// --------------------------------- halo ------------------------------------
static __device__ __forceinline__ float S_(const float* __restrict__ in,
                                           int b, int c, int j, int yy, int xx) {
  // in[b, j, c, yy, xx], faces dim=1, channels dim=2
  return in[((((size_t)b * 12 + j) * 32 + c) << 16) + (yy << 8) + xx];
}
#define S(j, yy, xx) S_(in, b, c, (j), (yy), (xx))

static __device__ float halo_value(const float* __restrict__ in,
                                   int b, int c, int face, int Y, int X) {
  const int i   = face & 3;
  const int cls = face >> 2;                  // 0=north 1=equatorial 2=south
  const int im1 = (i + 3) & 3, ip1 = (i + 1) & 3, ip2 = (i + 2) & 3;
  const bool top = (Y < P), bot = (Y >= P + N);
  const bool lft = (X < P), rgt = (X >= P + N);
  const int ym = Y - P, xm = X - P;           // middle-region coords
  const int yb = Y - P - N, xr = X - P - N;   // bottom/right local coords

  if (cls == 0) {                             // ---- northern polar face i
    if (top) {
      if (lft) return S(ip2, P - 1 - Y, P - 1 - X);        // rot(tl,2) corner
      if (rgt) return S(ip1, xr, P - 1 - Y);               // rot(tr,1)
      return S(ip1, xm, P - 1 - Y);                        // rot(t,1)
    }
    if (bot) {
      if (lft) return S(im1, yb, N - P + X);               // bl
      if (rgt) return S(8 + i, yb, xr);                    // br
      return S(4 + i, yb, xm);                             // b
    }
    if (lft) return S(im1, P - 1 - X, ym);                 // rot(l,-1)
    return S(4 + ip1, ym, xr);                             // r
  }
  if (cls == 1) {                             // ---- equatorial face 4+i
    if (top) {
      if (lft) {                                           // degenerate TL
        if (Y == P - 1 && X == P - 1)
          return 0.5f * S(i, N - 1, 0) + 0.5f * S(im1, 0, N - 1);
        if (X == P - 1) return S(i, N - P + Y, 0);
        if (Y == P - 1) return S(im1, 0, N - P + X);
        return 0.0f;
      }
      if (rgt) return S(i, N - P + Y, xr);                 // tr
      return S(i, N - P + Y, xm);                          // t
    }
    if (bot) {
      if (lft) return S(4 + im1, yb, N - P + X);           // bl
      if (rgt) {                                           // degenerate BR
        if (yb == 0 && xr == 0)
          return 0.5f * S(8 + i, 0, N - 1) + 0.5f * S(8 + im1, N - 1, 0);
        if (yb == 0) return S(8 + im1, N - 1, xr);
        if (xr == 0) return S(8 + i, yb, N - 1);
        return 0.0f;
      }
      return S(8 + i, yb, xm);                             // b
    }
    if (lft) return S(im1, ym, N - P + X);                 // l
    return S(4 + ip1, ym, xr);                             // r
  }
  // ---- southern polar face 8+i
  if (top) {
    if (lft) return S(4 + i, N - P + Y, X);                // tl[-p:, :p]
    if (rgt) return S(i, N - P + Y, N - P + xr);           // tr[-p:, -p:]
    return S(4 + i, N - P + Y, xm);                        // t
  }
  if (bot) {
    if (lft) return S(8 + ip1, N - P + X, N - 1 - yb);     // rot(bl,1)
    if (rgt) return S(8 + im1, N - 1 - yb, N - 1 - xr);    // rot(br,2)
    return S(8 + im1, xm, N - 1 - yb);                     // rot(b,1)
  }
  if (lft) return S(4 + ip1, ym, N - P + X);               // l
  return S(8 + ip1, N - 1 - xr, ym);                       // rot(r,-1)
}

__global__ __launch_bounds__(256)
void hp_halo(const float* __restrict__ in, float* __restrict__ out) {
  const int idx = blockIdx.x * 256 + threadIdx.x;   // halo element in frame
  if (idx >= 2 * HO * P + 2 * N * P) return;        // 2064
  const int pl  = blockIdx.y;
  const int b   = pl / (12 * 32);
  const int rem = pl - b * (12 * 32);
  const int face = rem >> 5;
  const int c    = rem & 31;

  int Y, X;
  if (idx < P * HO) {                 // top band (rows 0..1)
    Y = idx / HO; X = idx % HO;
  } else if (idx < 2 * P * HO) {      // bottom band (rows 258..259)
    const int q = idx - P * HO; Y = (P + N) + q / HO; X = q % HO;
  } else if (idx < 2 * P * HO + P * N) {  // left band
    const int q = idx - 2 * P * HO; Y = P + (q >> 1); X = q & 1;
  } else {                             // right band
    const int q = idx - (2 * P * HO + P * N); Y = P + (q >> 1); X = (P + N) + (q & 1);
  }
  out[(size_t)pl * (HO * HO) + (size_t)Y * HO + X] = halo_value(in, b, c, face, Y, X);
}

// ------------------------------- launcher ----------------------------------
extern "C" void kernel_launch(void* const* d_in, const int* in_sizes, int n_in,
                              void* d_out, int out_size, void* d_ws, size_t ws_size,
                              hipStream_t stream) {
  (void)n_in; (void)out_size; (void)d_ws; (void)ws_size;
  const float* in = (const float*)d_in[0];
  float* out = (float*)d_out;
  const int planes = in_sizes[0] >> 16;   // elements / (256*256) = B*12*C = 768

  dim3 gridI(N / ROWS, planes);           // (16, 768) blocks of 256
  hp_interior<<<gridI, 256, 0, stream>>>(in, out);

  const int haloN = 2 * HO * P + 2 * N * P;          // 2064
  dim3 gridH((haloN + 255) / 256, planes);           // (9, 768)
  hp_halo<<<gridH, 256, 0, stream>>>(in, out);
}